// ReconstructLoss_32959579029746
// MI455X (gfx1250) — compile-verified
//
#include <hip/hip_runtime.h>
#include <math.h>

typedef __attribute__((ext_vector_type(2))) float v2f;
typedef __attribute__((ext_vector_type(8))) float v8f;

#define NPTS     4096
#define EPS_F    1e-4f
#define L2E      1.4426950408889634f       /* log2(e) */
#define LN2      0.6931471805599453f
#define K2       (1.0e4f * L2E)            /* (1/eps) * log2(e): base-2 arg scale */
#define UNI_T    2.0f
#define NEG_BIG  (-3.0e38f)

#define BLOCK    512
#define WAVES    16                         /* BLOCK/32 */
#define TILES    16                         /* 4096 cols / WAVES / 16 */

// raw v_exp_f32 / v_log_f32 (base-2 transcendentals)
__device__ __forceinline__ float fexp2(float x) { return __builtin_amdgcn_exp2f(x); }
__device__ __forceinline__ float flog2(float x) { return __builtin_amdgcn_logf(x); }

// Merge two base-2 logsumexp states: (m,s) <- combine((m,s),(m2,s2)),
// state represents log2-sum = m + log2(s). Safe when one side is (-inf,0).
__device__ __forceinline__ void lse_merge(float& m, float& s, float m2, float s2) {
  float mn = fmaxf(m, m2);
  s = s * fexp2(m - mn) + s2 * fexp2(m2 - mn);
  m = mn;
}

// ---------------------------------------------------------------------------
// softmin over rows of C, C_ij = 0.5*(||r_i||^2 + ||c_j||^2) - <r_i, c_j>
//   f_i = -eps * ( LSE_j (h_j - C_ij)/eps - log(4096) )
// Base-2 domain: a2 = K2*(h_j - 0.5*cc_j) + K2*dot - 0.5*K2*rr_i
// symMode: out[i] = 0.5*(h[i] + f_i)
// grid: 256 blocks (16 rows each); block: 512 thr = 16 waves; wave: 16 tiles.
// LDS column pack: float4 (c0, c1, c2, hc).  B's K=3 row may be garbage (hc)
// since A's K=3 column is exactly zero.
// ---------------------------------------------------------------------------
__global__ __launch_bounds__(BLOCK)
void softmin_kernel(const float* __restrict__ rowPts, const float* __restrict__ colPts,
                    const float* __restrict__ rowSq,  const float* __restrict__ colSq,
                    const float* __restrict__ h, float* __restrict__ out, int symMode)
{
  __shared__ float4 colPack[NPTS];         // (c0, c1, c2, K2*(h - 0.5*cc))
  __shared__ float pm[WAVES][16], ps[WAVES][16];

  const int tid   = threadIdx.x;
  const int lane  = tid & 31;
  const int wave  = tid >> 5;
  const int laneM = lane & 15;
  const int hiH   = lane >> 4;             // 0: K=0,1   1: K=2,pad
  const int rowBase = blockIdx.x << 4;

  for (int j = tid; j < NPTS; j += BLOCK) {
    const float* yj = colPts + (size_t)j * 3;
    float4 q;
    q.x = yj[0]; q.y = yj[1]; q.z = yj[2];
    q.w = K2 * __builtin_fmaf(-0.5f, colSq[j], h[j]);
    colPack[j] = q;
  }

  // A operand: 16x4 f32, lane L holds row M=L%16, K pair (0,1) or (2,3=0)
  v2f a;
  {
    const float* xr = rowPts + (size_t)(rowBase + laneM) * 3;
    a[0] = hiH ? xr[2] : xr[0];
    a[1] = hiH ? 0.0f  : xr[1];
  }
  float rt[8];
#pragma unroll
  for (int v = 0; v < 8; ++v)
    rt[v] = -0.5f * K2 * rowSq[rowBase + v + 8 * hiH];

  float m[8], s[8];
#pragma unroll
  for (int v = 0; v < 8; ++v) { m[v] = -__builtin_inff(); s[v] = 0.0f; }

  __syncthreads();

  const int colStart = wave * (TILES * 16);
  const float* cp   = (const float*)colPack;
  const float* bptr = cp + (size_t)(colStart + laneM) * 4 + hiH * 2;
  const float* hptr = cp + (size_t)(colStart + laneM) * 4 + 3;

  for (int t = 0; t < TILES; ++t) {
    v2f b = *(const v2f*)(bptr + t * 64);   // B: 4x16, lane L holds col N=L%16
    const float hcv = hptr[t * 64];
    v8f acc = {};
    acc = __builtin_amdgcn_wmma_f32_16x16x4_f32(false, a, false, b,
                                                (short)0, acc, false, false);
#pragma unroll
    for (int v = 0; v < 8; ++v) {
      float a2 = __builtin_fmaf(K2, acc[v], hcv + rt[v]);   // base-2 arg
      float mn = fmaxf(m[v], a2);
      s[v] = __builtin_fmaf(s[v], fexp2(m[v] - mn), fexp2(a2 - mn));
      m[v] = mn;
    }
  }

  // reduce across the 16 lanes holding the same row (xor within half-wave)
#pragma unroll
  for (int d = 1; d <= 8; d <<= 1) {
#pragma unroll
    for (int v = 0; v < 8; ++v) {
      float m2 = __shfl_xor(m[v], d, 32);
      float s2 = __shfl_xor(s[v], d, 32);
      lse_merge(m[v], s[v], m2, s2);
    }
  }

  if (lane == 0) {
#pragma unroll
    for (int v = 0; v < 8; ++v) { pm[wave][v] = m[v]; ps[wave][v] = s[v]; }
  }
  if (lane == 16) {
#pragma unroll
    for (int v = 0; v < 8; ++v) { pm[wave][8 + v] = m[v]; ps[wave][8 + v] = s[v]; }
  }
  __syncthreads();

  if (tid < 16) {
    const int r = tid;
    float M = pm[0][r], S = ps[0][r];
#pragma unroll
    for (int w = 1; w < WAVES; ++w) lse_merge(M, S, pm[w][r], ps[w][r]);
    // lse (natural log) = LN2*(M + log2(S)); log(4096) = LN2*12
    float f = -EPS_F * LN2 * (M + flog2(S) - 12.0f);
    out[rowBase + r] = symMode ? 0.5f * (h[rowBase + r] + f) : f;
  }
}

// ---------------------------------------------------------------------------
// uniformity partials per block over 16 rows x 4096 cols of
//   arg = -UNI_T * ||x_i - x_j||^2  (diag masked), base-2 state out.
// a2 = (-2*L2E)*cc + (4*L2E)*dot + (-2*L2E)*rr
// ---------------------------------------------------------------------------
__global__ __launch_bounds__(BLOCK)
void uniformity_kernel(const float* __restrict__ pts, const float* __restrict__ sq,
                       float* __restrict__ um, float* __restrict__ us)
{
  __shared__ float4 colPack[NPTS];         // (x0, x1, x2, -UNI_T*L2E*sq)
  __shared__ float pm[WAVES][16], ps[WAVES][16];
  __shared__ float rmM[16], rmS[16];

  const int tid   = threadIdx.x;
  const int lane  = tid & 31;
  const int wave  = tid >> 5;
  const int laneM = lane & 15;
  const int hiH   = lane >> 4;
  const int rowBase = blockIdx.x << 4;

  for (int j = tid; j < NPTS; j += BLOCK) {
    const float* xj = pts + (size_t)j * 3;
    float4 q;
    q.x = xj[0]; q.y = xj[1]; q.z = xj[2];
    q.w = (-UNI_T * L2E) * sq[j];
    colPack[j] = q;
  }

  v2f a;
  {
    const float* xr = pts + (size_t)(rowBase + laneM) * 3;
    a[0] = hiH ? xr[2] : xr[0];
    a[1] = hiH ? 0.0f  : xr[1];
  }
  float rt[8];
  int rowIdx[8];
#pragma unroll
  for (int v = 0; v < 8; ++v) {
    rowIdx[v] = rowBase + v + 8 * hiH;
    rt[v] = (-UNI_T * L2E) * sq[rowIdx[v]];
  }

  float m[8], s[8];
#pragma unroll
  for (int v = 0; v < 8; ++v) { m[v] = -__builtin_inff(); s[v] = 0.0f; }

  __syncthreads();

  const int colStart = wave * (TILES * 16);
  const float* cp   = (const float*)colPack;
  const float* bptr = cp + (size_t)(colStart + laneM) * 4 + hiH * 2;
  const float* hptr = cp + (size_t)(colStart + laneM) * 4 + 3;

  for (int t = 0; t < TILES; ++t) {
    const int col = colStart + (t << 4) + laneM;
    v2f b = *(const v2f*)(bptr + t * 64);
    const float hcv = hptr[t * 64];
    v8f acc = {};
    acc = __builtin_amdgcn_wmma_f32_16x16x4_f32(false, a, false, b,
                                                (short)0, acc, false, false);
#pragma unroll
    for (int v = 0; v < 8; ++v) {
      float a2 = __builtin_fmaf(2.0f * UNI_T * L2E, acc[v], hcv + rt[v]);
      if (rowIdx[v] == col) a2 = NEG_BIG;                  // mask self-pair
      float mn = fmaxf(m[v], a2);
      s[v] = __builtin_fmaf(s[v], fexp2(m[v] - mn), fexp2(a2 - mn));
      m[v] = mn;
    }
  }

#pragma unroll
  for (int d = 1; d <= 8; d <<= 1) {
#pragma unroll
    for (int v = 0; v < 8; ++v) {
      float m2 = __shfl_xor(m[v], d, 32);
      float s2 = __shfl_xor(s[v], d, 32);
      lse_merge(m[v], s[v], m2, s2);
    }
  }

  if (lane == 0) {
#pragma unroll
    for (int v = 0; v < 8; ++v) { pm[wave][v] = m[v]; ps[wave][v] = s[v]; }
  }
  if (lane == 16) {
#pragma unroll
    for (int v = 0; v < 8; ++v) { pm[wave][8 + v] = m[v]; ps[wave][8 + v] = s[v]; }
  }
  __syncthreads();
  if (tid < 16) {
    const int r = tid;
    float M = pm[0][r], S = ps[0][r];
#pragma unroll
    for (int w = 1; w < WAVES; ++w) lse_merge(M, S, pm[w][r], ps[w][r]);
    rmM[r] = M; rmS[r] = S;
  }
  __syncthreads();
  if (tid == 0) {
    float M = rmM[0], S = rmS[0];
    for (int r = 1; r < 16; ++r) lse_merge(M, S, rmM[r], rmS[r]);
    um[blockIdx.x] = M;                     // base-2 state
    us[blockIdx.x] = S;
  }
}

// ---------------------------------------------------------------------------
__global__ __launch_bounds__(256)
void prep_kernel(const float* __restrict__ pq, const float* __restrict__ pgt,
                 float* __restrict__ yc, float* __restrict__ xsq, float* __restrict__ ysq,
                 float* __restrict__ f, float* __restrict__ g,
                 float* __restrict__ px, float* __restrict__ py)
{
  int j = blockIdx.x * 256 + threadIdx.x;
  if (j >= NPTS) return;
  float x0 = pq[j * 3 + 0], x1 = pq[j * 3 + 1], x2 = pq[j * 3 + 2];
  xsq[j] = x0 * x0 + x1 * x1 + x2 * x2;
  float y0 = pgt[j * 4 + 0], y1 = pgt[j * 4 + 1], y2 = pgt[j * 4 + 2];
  yc[j * 3 + 0] = y0; yc[j * 3 + 1] = y1; yc[j * 3 + 2] = y2;
  ysq[j] = y0 * y0 + y1 * y1 + y2 * y2;
  f[j] = 0.0f; g[j] = 0.0f; px[j] = 0.0f; py[j] = 0.0f;
}

// ---------------------------------------------------------------------------
__global__ __launch_bounds__(256)
void finalize_kernel(const float* __restrict__ fF, const float* __restrict__ gF,
                     const float* __restrict__ fx, const float* __restrict__ fy,
                     const float* __restrict__ um, const float* __restrict__ us,
                     float* __restrict__ out)
{
  __shared__ float rsum[256];
  __shared__ float rm[256], rs[256];
  int tid = threadIdx.x;
  float acc = 0.0f;
  for (int i = tid; i < NPTS; i += 256)
    acc += fF[i] + gF[i] - fx[i] - fy[i];
  rsum[tid] = acc;
  rm[tid] = um[tid];
  rs[tid] = us[tid];
  __syncthreads();
  for (int step = 128; step > 0; step >>= 1) {
    if (tid < step) {
      rsum[tid] += rsum[tid + step];
      lse_merge(rm[tid], rs[tid], rm[tid + step], rs[tid + step]);
    }
    __syncthreads();
  }
  if (tid == 0) {
    float U = LN2 * (rm[0] + flog2(rs[0])) - __logf((float)NPTS * (float)(NPTS - 1));
    out[0] = rsum[0] * (1.0f / (float)NPTS) + U;
  }
}

// ---------------------------------------------------------------------------
extern "C" void kernel_launch(void* const* d_in, const int* in_sizes, int n_in,
                              void* d_out, int out_size, void* d_ws, size_t ws_size,
                              hipStream_t stream) {
  (void)in_sizes; (void)n_in; (void)out_size; (void)ws_size;
  const float* x   = (const float*)d_in[0];   // [4096,3]
  const float* pgt = (const float*)d_in[1];   // [4096,4]
  float* ws = (float*)d_ws;

  float* yc  = ws;                    // 4096*3
  float* xsq = yc  + NPTS * 3;
  float* ysq = xsq + NPTS;
  float* f   = ysq + NPTS;
  float* g   = f   + NPTS;
  float* fF  = g   + NPTS;
  float* gF  = fF  + NPTS;
  float* px  = gF  + NPTS;
  float* pxb = px  + NPTS;
  float* py  = pxb + NPTS;
  float* pyb = py  + NPTS;
  float* fx  = pyb + NPTS;
  float* fy  = fx  + NPTS;
  float* um  = fy  + NPTS;            // 256
  float* us  = um  + 256;             // 256

  dim3 gridS(256), blockS(BLOCK);

  prep_kernel<<<16, 256, 0, stream>>>(x, pgt, yc, xsq, ysq, f, g, px, py);

  // Sinkhorn OT(x,y): 20 alternating iterations
  for (int it = 0; it < 20; ++it) {
    softmin_kernel<<<gridS, blockS, 0, stream>>>(x,  yc, xsq, ysq, g, f, 0);
    softmin_kernel<<<gridS, blockS, 0, stream>>>(yc, x,  ysq, xsq, f, g, 0);
  }
  softmin_kernel<<<gridS, blockS, 0, stream>>>(x,  yc, xsq, ysq, g, fF, 0);
  softmin_kernel<<<gridS, blockS, 0, stream>>>(yc, x,  ysq, xsq, f, gF, 0);

  // symmetric debias fixed points (ping-pong)
  {
    float* cur = px; float* nxt = pxb;
    for (int it = 0; it < 20; ++it) {
      softmin_kernel<<<gridS, blockS, 0, stream>>>(x, x, xsq, xsq, cur, nxt, 1);
      float* t = cur; cur = nxt; nxt = t;
    }
    softmin_kernel<<<gridS, blockS, 0, stream>>>(x, x, xsq, xsq, cur, fx, 0);
  }
  {
    float* cur = py; float* nxt = pyb;
    for (int it = 0; it < 20; ++it) {
      softmin_kernel<<<gridS, blockS, 0, stream>>>(yc, yc, ysq, ysq, cur, nxt, 1);
      float* t = cur; cur = nxt; nxt = t;
    }
    softmin_kernel<<<gridS, blockS, 0, stream>>>(yc, yc, ysq, ysq, cur, fy, 0);
  }

  uniformity_kernel<<<gridS, blockS, 0, stream>>>(x, xsq, um, us);

  finalize_kernel<<<1, 256, 0, stream>>>(fF, gF, fx, fy, um, us, (float*)d_out);
}